// GCN_40312563040379
// MI455X (gfx1250) — compile-verified
//
#include <hip/hip_runtime.h>

typedef __attribute__((ext_vector_type(2))) float v2f;
typedef __attribute__((ext_vector_type(8))) float v8f;

#define GCN_D 128

// ---------------------------------------------------------------------------
// Dense projection: C[nRows x 128] = X[nRows x 128] @ W[128 x 128]
// One wave per 16-row tile; 8 column tiles of 16; K looped in steps of 4
// using V_WMMA_F32_16X16X4_F32 (full fp32 precision).
// A frag (ISA 7.12.2, 32-bit A 16x4): lane m = lane%16, k = 2*(lane/16)+v
// C/D frag (32-bit 16x16):            VGPR r -> M = r + 8*(lane/16), N = lane%16
// ---------------------------------------------------------------------------
__global__ __launch_bounds__(256) void gcn_gemm_wmma(
    const float* __restrict__ X, const float* __restrict__ W,
    float* __restrict__ C, int nRows) {
  const int wave  = (int)((blockIdx.x * blockDim.x + threadIdx.x) >> 5);
  const int lane  = (int)(threadIdx.x & 31);
  const int nTile = nRows >> 4;               // nRows is a multiple of 16
  if (wave >= nTile) return;                  // uniform per-wave exit: EXEC stays all-1s

  const int m0    = wave << 4;
  const int mrow  = lane & 15;
  const int khalf = (lane >> 4) << 1;         // 0 or 2

  v8f acc[8];
#pragma unroll
  for (int t = 0; t < 8; ++t)
    acc[t] = (v8f){0.f, 0.f, 0.f, 0.f, 0.f, 0.f, 0.f, 0.f};

  const float* xrow = X + (size_t)(m0 + mrow) * GCN_D;

#pragma unroll 4
  for (int k0 = 0; k0 < GCN_D; k0 += 4) {
    // A fragment: float2 = X[m, k0+khalf .. k0+khalf+1]
    v2f a = *(const v2f*)(xrow + k0 + khalf);
#pragma unroll
    for (int t = 0; t < 8; ++t) {
      // B fragment: W[k0+khalf+v, t*16 + mrow], v = 0,1
      const float* wp = W + (size_t)(k0 + khalf) * GCN_D + t * 16 + mrow;
      v2f b;
      b.x = wp[0];
      b.y = wp[GCN_D];
      acc[t] = __builtin_amdgcn_wmma_f32_16x16x4_f32(
          /*neg_a=*/false, a, /*neg_b=*/false, b,
          /*c_mod=*/(short)0, acc[t], /*reuse_a=*/false, /*reuse_b=*/false);
    }
  }

  const int mtop = (lane >> 4) << 3;          // 0 or 8
#pragma unroll
  for (int t = 0; t < 8; ++t) {
#pragma unroll
    for (int r = 0; r < 8; ++r) {
      C[(size_t)(m0 + mtop + r) * GCN_D + t * 16 + mrow] = acc[t][r];
    }
  }
}

// ---------------------------------------------------------------------------
// Edge scatter: agg[dst] += w_e * support[src].  One wave per edge; each lane
// handles 4 consecutive floats (512B coalesced read per wave from L2-resident
// support), then 4x global_atomic_add_f32 into the L2-resident accumulator.
// ---------------------------------------------------------------------------
__global__ __launch_bounds__(256) void gcn_edge_scatter(
    const float* __restrict__ support,
    const int* __restrict__ esrc, const int* __restrict__ edst,
    const float* __restrict__ ew,
    float* __restrict__ agg, int nEdges) {
  const int edge = (int)((blockIdx.x * blockDim.x + threadIdx.x) >> 5);
  const int lane = (int)(threadIdx.x & 31);
  if (edge >= nEdges) return;

  const int   s = esrc[edge];
  const int   d = edst[edge];
  const float w = ew[edge];

  const float4 v = *((const float4*)(support + (size_t)s * GCN_D) + lane);
  float* op = agg + (size_t)d * GCN_D + lane * 4;
  atomicAdd(op + 0, v.x * w);
  atomicAdd(op + 1, v.y * w);
  atomicAdd(op + 2, v.z * w);
  atomicAdd(op + 3, v.w * w);
}

// ---------------------------------------------------------------------------
// Elementwise helpers
// ---------------------------------------------------------------------------
__global__ __launch_bounds__(256) void gcn_zero_f4(float4* __restrict__ p, int n4) {
  const int i = (int)(blockIdx.x * blockDim.x + threadIdx.x);
  if (i < n4) p[i] = make_float4(0.f, 0.f, 0.f, 0.f);
}

// In-place: h = (relu ? max(0, h + b) : h + b), float4 per thread.
// 128 floats/row = 32 float4 -> bias float4 index = i & 31.
__global__ __launch_bounds__(256) void gcn_bias_act(
    float4* __restrict__ h, const float4* __restrict__ b4, int n4, int doRelu) {
  const int i = (int)(blockIdx.x * blockDim.x + threadIdx.x);
  if (i >= n4) return;
  const float4 b = b4[i & 31];
  float4 v = h[i];
  v.x += b.x; v.y += b.y; v.z += b.z; v.w += b.w;
  if (doRelu) {
    v.x = fmaxf(v.x, 0.f); v.y = fmaxf(v.y, 0.f);
    v.z = fmaxf(v.z, 0.f); v.w = fmaxf(v.w, 0.f);
  }
  h[i] = v;
}

// ---------------------------------------------------------------------------
// Launch
// ---------------------------------------------------------------------------
extern "C" void kernel_launch(void* const* d_in, const int* in_sizes, int n_in,
                              void* d_out, int out_size, void* d_ws, size_t ws_size,
                              hipStream_t stream) {
  const float* features = (const float*)d_in[0];
  const int*   edge_src = (const int*)d_in[1];
  const int*   edge_dst = (const int*)d_in[2];
  const float* edge_w   = (const float*)d_in[3];
  const float* W1       = (const float*)d_in[4];
  const float* b1       = (const float*)d_in[5];
  const float* W2       = (const float*)d_in[6];
  const float* b2       = (const float*)d_in[7];

  const int nNodes = in_sizes[0] / GCN_D;     // 50000
  const int nEdges = in_sizes[1];             // 1600000
  const int nElem  = nNodes * GCN_D;          // 6.4M floats
  const int n4     = nElem / 4;               // 1.6M float4

  float* bufA = (float*)d_ws;                 // projection result (support)
  float* bufB = bufA + (size_t)nElem;         // aggregation / hidden

  float* out = (float*)d_out;

  const int TPB = 256;
  dim3 blk(TPB);
  dim3 gElem((n4 + TPB - 1) / TPB);                           // elementwise grids
  dim3 gGemm(((nNodes / 16) + (TPB / 32) - 1) / (TPB / 32));  // 1 wave / 16 rows
  dim3 gEdge((nEdges + (TPB / 32) - 1) / (TPB / 32));         // 1 wave / edge

  // ---- Layer 1: h = relu( scatter(edges, X@W1) + b1 ) ----
  gcn_zero_f4<<<gElem, blk, 0, stream>>>((float4*)bufB, n4);
  gcn_gemm_wmma<<<gGemm, blk, 0, stream>>>(features, W1, bufA, nNodes);
  gcn_edge_scatter<<<gEdge, blk, 0, stream>>>(bufA, edge_src, edge_dst, edge_w,
                                              bufB, nEdges);
  gcn_bias_act<<<gElem, blk, 0, stream>>>((float4*)bufB, (const float4*)b1, n4, 1);

  // ---- Layer 2: out = scatter(edges, h@W2) + b2 ----
  gcn_gemm_wmma<<<gGemm, blk, 0, stream>>>(bufB, W2, bufA, nNodes);
  gcn_zero_f4<<<gElem, blk, 0, stream>>>((float4*)out, n4);
  gcn_edge_scatter<<<gEdge, blk, 0, stream>>>(bufA, edge_src, edge_dst, edge_w,
                                              out, nEdges);
  gcn_bias_act<<<gElem, blk, 0, stream>>>((float4*)out, (const float4*)b2, n4, 0);
}